// CompiledModel_18751827215057
// MI455X (gfx1250) — compile-verified
//
#include <hip/hip_runtime.h>
#include <hip/hip_bf16.h>

typedef float v2f __attribute__((ext_vector_type(2)));
typedef float v4f __attribute__((ext_vector_type(4)));
typedef float v8f __attribute__((ext_vector_type(8)));
typedef int   v4i __attribute__((ext_vector_type(4)));

#define ROW_PITCH 68        // 64 floats + 4 pad -> conflict-free LDS bank map
#define WAVES_PER_BLOCK 4
#define BLOCK_THREADS (WAVES_PER_BLOCK * 32)
#define TILE_FLOATS (16 * ROW_PITCH)

#if __has_builtin(__builtin_amdgcn_global_load_async_to_lds_b128)
#define USE_ASYNC 1
#else
#define USE_ASYNC 0
#endif

// ---------------------------------------------------------------------------
// Kernel 1: fold query through WQ/bQ/WK into per-head key vectors; init winners
// kvec[h][d] = (WQ[h,0].q + bQ[h,0]) * WK[h,0,d] + (WQ[h,1].q + bQ[h,1]) * WK[h,1,d]
// ---------------------------------------------------------------------------
__global__ __launch_bounds__(256) void prep_kernel(
    const float* __restrict__ query, const float* __restrict__ WQ,
    const float* __restrict__ bQ, const float* __restrict__ WK,
    float* __restrict__ kvec, unsigned long long* __restrict__ winners)
{
    for (int idx = threadIdx.x; idx < 640; idx += blockDim.x) {
        const int h = idx >> 6, d = idx & 63;
        float q0 = bQ[h * 2 + 0], q1 = bQ[h * 2 + 1];
        for (int j = 0; j < 64; ++j) {
            q0 += WQ[(h * 2 + 0) * 64 + j] * query[j];
            q1 += WQ[(h * 2 + 1) * 64 + j] * query[j];
        }
        kvec[h * 64 + d] = q0 * WK[(h * 2 + 0) * 64 + d]
                         + q1 * WK[(h * 2 + 1) * 64 + d];
    }
    if (threadIdx.x < 10) winners[threadIdx.x] = 0ull;
}

// ---------------------------------------------------------------------------
// Kernel 2: stream one bank, score all its heads with V_WMMA_F32_16X16X4_F32,
// block-reduce argmax, one packed u64 atomicMax per head per block.
// Double-buffered async global->LDS staging when the toolchain exposes it.
// ---------------------------------------------------------------------------
__global__ __launch_bounds__(BLOCK_THREADS) void scan_bank_kernel(
    const float* __restrict__ mem, int nrows,
    const float* __restrict__ kvec,
    unsigned long long* __restrict__ winners,
    int head_base, int nheads)
{
#if USE_ASYNC
    __shared__ float lds_tile[WAVES_PER_BLOCK * 2 * TILE_FLOATS];
#else
    __shared__ float lds_tile[WAVES_PER_BLOCK * TILE_FLOATS];
#endif
    __shared__ unsigned long long red[WAVES_PER_BLOCK * 2 * 3];

    const int lane  = threadIdx.x & 31;
    const int wave  = threadIdx.x >> 5;
    const int n     = lane & 15;      // N column of D / M row of A
    const int hi    = lane >> 4;      // selects K pair {0,1} vs {2,3}
    const int nwaves = (blockDim.x >> 5) * gridDim.x;
    const int gwave  = blockIdx.x * (blockDim.x >> 5) + wave;

    // B fragments: B[k][col] = kvec[head col][kk*4 + k]; columns >= nheads are 0.
    v2f bfrag[16];
#pragma unroll
    for (int kk = 0; kk < 16; ++kk) {
        float b0 = 0.0f, b1 = 0.0f;
        if (n < nheads) {
            const float* kv = kvec + (size_t)(head_base + n) * 64 + kk * 4 + hi * 2;
            b0 = kv[0];
            b1 = kv[1];
        }
        v2f b; b.x = b0; b.y = b1;
        bfrag[kk] = b;
    }

    unsigned long long best = 0ull;   // packed {orderable f32 key, ~row}
    const int ntiles = nrows >> 4;

#define ARGMAX_UPDATE(ACC, T)                                                  \
    if (n < nheads) {                                                          \
        _Pragma("unroll") for (int r = 0; r < 8; ++r) {                        \
            const float s = (ACC)[r];                                          \
            const unsigned int row = (unsigned int)((T) * 16 + r + hi * 8);    \
            const unsigned int u = __float_as_uint(s);                         \
            const unsigned int key =                                           \
                (u & 0x80000000u) ? ~u : (u | 0x80000000u);                    \
            const unsigned long long p =                                       \
                ((unsigned long long)key << 32) | (unsigned int)(~row);        \
            if (p > best) best = p;                                            \
        }                                                                      \
    }

#if USE_ASYNC
    float* buf0 = &lds_tile[wave * 2 * TILE_FLOATS];
    float* buf1 = buf0 + TILE_FLOATS;

    auto issue = [&](int t, float* buf) {
        const float* src = mem + (size_t)t * 16 * 64;
#pragma unroll
        for (int i = 0; i < 8; ++i) {
            const int e = i * 32 + lane;          // 16-byte unit 0..255
            const int r = e >> 4;
            const int c = (e & 15) * 4;
            __builtin_amdgcn_global_load_async_to_lds_b128(
                (__attribute__((address_space(1))) v4i*)(src + (size_t)e * 4),
                (__attribute__((address_space(3))) v4i*)(buf + r * ROW_PITCH + c),
                0, 0);
        }
    };

    int t = gwave;
    int par = 0;
    if (t < ntiles) issue(t, buf0);
    for (; t < ntiles; t += nwaves) {
        float* cur = par ? buf1 : buf0;
        float* nxt = par ? buf0 : buf1;
        const int tn = t + nwaves;
        if (tn < ntiles) {
            issue(tn, nxt);                        // prefetch next tile
            asm volatile("s_wait_asynccnt 0x8" ::: "memory");  // cur landed
        } else {
            asm volatile("s_wait_asynccnt 0x0" ::: "memory");
        }
        v8f acc = (v8f)0.0f;
#pragma unroll
        for (int kk = 0; kk < 16; ++kk) {
            // A fragment: lane holds row M=n, K = kk*4 + hi*2 + {0,1}
            v2f a = *(const v2f*)(cur + n * ROW_PITCH + kk * 4 + hi * 2);
            acc = __builtin_amdgcn_wmma_f32_16x16x4_f32(
                false, a, false, bfrag[kk], (short)0, acc, false, false);
        }
        ARGMAX_UPDATE(acc, t)
        par ^= 1;
    }
#else
    float* tile = &lds_tile[wave * TILE_FLOATS];
    for (int t = gwave; t < ntiles; t += nwaves) {
        const float* src = mem + (size_t)t * 16 * 64;
        // Stage 16x64 tile into padded LDS: coalesced non-temporal b128 loads.
#pragma unroll
        for (int i = 0; i < 8; ++i) {
            const int e = i * 32 + lane;
            const int r = e >> 4;
            const int c = (e & 15) * 4;
            v4f v = __builtin_nontemporal_load((const v4f*)(src + (size_t)e * 4));
            *(v4f*)(tile + r * ROW_PITCH + c) = v;
        }
        asm volatile("s_wait_dscnt 0" ::: "memory");  // same-wave LDS RAW
        v8f acc = (v8f)0.0f;
#pragma unroll
        for (int kk = 0; kk < 16; ++kk) {
            v2f a = *(const v2f*)(tile + n * ROW_PITCH + kk * 4 + hi * 2);
            acc = __builtin_amdgcn_wmma_f32_16x16x4_f32(
                false, a, false, bfrag[kk], (short)0, acc, false, false);
        }
        ARGMAX_UPDATE(acc, t)
    }
#endif

    if (n < nheads) red[(wave * 2 + hi) * 3 + n] = best;
    __syncthreads();

    if ((int)threadIdx.x < nheads) {
        unsigned long long b = 0ull;
        for (int s = 0; s < WAVES_PER_BLOCK * 2; ++s) {
            const unsigned long long c = red[s * 3 + threadIdx.x];
            if (c > b) b = c;
        }
        atomicMax(&winners[head_base + threadIdx.x], b);
    }
}

// ---------------------------------------------------------------------------
// Kernel 3: decode winners, gather mem[best], apply WV, pack outputs.
// d_out: [0..11] values (f32), [12..21] best scores (f32), [22..31] idx (i32)
// ---------------------------------------------------------------------------
__global__ __launch_bounds__(32) void finalize_kernel(
    const float* __restrict__ prog,  const float* __restrict__ stack_,
    const float* __restrict__ local_, const float* __restrict__ heap,
    const float* __restrict__ call,
    const float* __restrict__ WV1,   const float* __restrict__ WVc,
    const unsigned long long* __restrict__ winners, float* __restrict__ out)
{
    const int h = threadIdx.x;
    if (h >= 10) return;

    const float* mem;
    if (h < 2)      mem = prog;
    else if (h < 5) mem = stack_;
    else if (h < 7) mem = local_;
    else if (h < 9) mem = heap;
    else            mem = call;

    const unsigned long long w = winners[h];
    const unsigned int key = (unsigned int)(w >> 32);
    const unsigned int row = ~(unsigned int)(w & 0xFFFFFFFFull);
    const unsigned int u = (key & 0x80000000u) ? (key & 0x7FFFFFFFu) : ~key;
    const float score = __uint_as_float(u);

    const float* m = mem + (size_t)row * 64;
    if (h < 9) {
        float v = 0.0f;
        for (int j = 0; j < 64; ++j) v += WV1[h * 64 + j] * m[j];
        out[h] = v;
    } else {
        for (int t = 0; t < 3; ++t) {
            float v = 0.0f;
            for (int j = 0; j < 64; ++j) v += WVc[t * 64 + j] * m[j];
            out[9 + t] = v;
        }
    }
    out[12 + h] = score;
    ((int*)out)[22 + h] = (int)row;
}

// ---------------------------------------------------------------------------
extern "C" void kernel_launch(void* const* d_in, const int* in_sizes, int n_in,
                              void* d_out, int out_size, void* d_ws, size_t ws_size,
                              hipStream_t stream)
{
    (void)in_sizes; (void)n_in; (void)out_size; (void)ws_size;

    const float* query  = (const float*)d_in[0];
    const float* prog   = (const float*)d_in[1];   // 262144 x 64
    const float* stack_ = (const float*)d_in[2];   // 131072 x 64
    const float* local_ = (const float*)d_in[3];   // 131072 x 64
    const float* heap   = (const float*)d_in[4];   // 262144 x 64
    const float* call   = (const float*)d_in[5];   //  32768 x 64
    const float* WQ     = (const float*)d_in[6];
    const float* bQ     = (const float*)d_in[7];
    const float* WK     = (const float*)d_in[8];
    const float* WV1    = (const float*)d_in[9];
    const float* WVc    = (const float*)d_in[10];

    float* kvec = (float*)d_ws;                                       // 10*64 f32
    unsigned long long* winners =
        (unsigned long long*)((char*)d_ws + 10 * 64 * sizeof(float)); // 10 u64

    prep_kernel<<<1, 256, 0, stream>>>(query, WQ, bQ, WK, kvec, winners);

    auto scan = [&](const float* mem, int nrows, int head_base, int nheads) {
        const int tiles = nrows >> 4;
        int blocks = (tiles + WAVES_PER_BLOCK - 1) / WAVES_PER_BLOCK;
        if (blocks > 1024) blocks = 1024;
        scan_bank_kernel<<<blocks, BLOCK_THREADS, 0, stream>>>(
            mem, nrows, kvec, winners, head_base, nheads);
    };
    scan(prog,   262144, 0, 2);
    scan(stack_, 131072, 2, 3);
    scan(local_, 131072, 5, 2);
    scan(heap,   262144, 7, 2);
    scan(call,    32768, 9, 1);

    finalize_kernel<<<1, 32, 0, stream>>>(prog, stack_, local_, heap, call,
                                          WV1, WVc, winners, (float*)d_out);
}